// OCRTorchScript_47820165874441
// MI455X (gfx1250) — compile-verified
//
#include <hip/hip_runtime.h>

typedef __attribute__((ext_vector_type(2))) float v2f;
typedef __attribute__((ext_vector_type(8))) float v8f;

#define IMG_H   1080
#define IMG_W   1920
#define CROP_H  80
#define CROP_W  640
#define OUT_H   40
#define OUT_W   320

// Exact 2x bicubic (a = -0.75) taps at distances {1.5, 0.5, 0.5, 1.5}
#define F_OUTER (-0.09375f)
#define F_INNER ( 0.59375f)

#if __has_builtin(__builtin_amdgcn_global_load_async_to_lds_b32)
#define HAVE_ASYNC_LDS 1
typedef __attribute__((address_space(1))) int gas_int;   // global (AS1) int
typedef __attribute__((address_space(3))) int las_int;   // LDS (AS3) int
#endif

// Branchless tap weight: d in band [0,4) -> {F_OUTER,F_INNER,F_INNER,F_OUTER}, else 0
__device__ __forceinline__ float wsel(int d) {
  const float w = (d == 1 || d == 2) ? F_INNER : F_OUTER;
  return ((unsigned)d < 4u) ? w : 0.0f;
}

// H-pass weight matrix Wh[o][r]; rows o >= 40 are zero padding.
// Row o taps input rows 2o-1 .. 2o+2; clamped taps fold into rows 0 / 79.
__device__ __forceinline__ float whval(int o, int r) {
  float w = wsel(r - (2 * o - 1));
  w = (o < OUT_H) ? w : 0.0f;
  w += ((o == 0) & (r == 0)) ? F_OUTER : 0.0f;
  w += ((o == OUT_H - 1) & (r == CROP_H - 1)) ? F_OUTER : 0.0f;
  return w;
}

#define IN_S   81   // padded LDS row strides (mod-64 offsets conflict-free)
#define MID_S  81

__global__ void __launch_bounds__(256)
crop_resize_bicubic_wmma(const float* __restrict__ img,
                         const int*   __restrict__ boxes,
                         float*       __restrict__ out)
{
  __shared__ float sIn[CROP_H][IN_S];   // 80 x 81 fp32: one-channel input strip
  __shared__ float sMid[48][MID_S];     // 48 x 81 fp32: H-resized strip

  const int strip = blockIdx.x / 3;     // 0..9, 32 output columns each
  const int ch    = blockIdx.x % 3;
  const int box   = blockIdx.y;

  const int x0 = boxes[box * 5 + 0];
  const int y0 = boxes[box * 5 + 1];
  const int bi = boxes[box * 5 + 4];

  const int tid   = threadIdx.x;
  // Scalarize the wave index: makes all tile/K loop bounds SGPR-uniform so the
  // compiler emits plain scalar loops and EXEC stays all-ones at every WMMA.
  const int wave  = __builtin_amdgcn_readfirstlane(tid >> 5);
  const int lane  = tid & 31;
  const int lrow  = lane & 15;
  const int lhalf = lane >> 4;

  // ---- stage 0: load 80x80 crop window (8-col halo each side, replicate) ----
  // LDS col j <-> crop col (64*strip - 8 + j), clamped at load time, which
  // realizes border replication for BOTH resize passes. Uses CDNA5 async
  // global->LDS gather (per-lane addresses, no VGPR round-trip).
  const int cbase = strip * 64 - 8;
  for (int idx = tid; idx < CROP_H * 80; idx += 256) {
    const int r = idx / 80;
    const int j = idx - r * 80;
    int gc = cbase + j;
    gc = gc < 0 ? 0 : (gc > CROP_W - 1 ? CROP_W - 1 : gc);
    const size_t src = ((size_t)(bi * IMG_H + (y0 + r)) * IMG_W + (x0 + gc)) * 3 + ch;
#if HAVE_ASYNC_LDS
    __builtin_amdgcn_global_load_async_to_lds_b32(
        (gas_int*)(img + src), (las_int*)&sIn[r][j], 0, 0);
#else
    sIn[r][j] = img[src];
#endif
  }
#if HAVE_ASYNC_LDS
#if __has_builtin(__builtin_amdgcn_s_wait_asynccnt)
  __builtin_amdgcn_s_wait_asynccnt(0);
#else
  asm volatile("s_wait_asynccnt 0x0" ::: "memory");
#endif
#endif
  __syncthreads();

  // ---- stage 1: sMid[48][80] = Wh[48][80] @ sIn[80][80], banded K ----
  // Every M-tile uses exactly 10 K-steps; out-of-band extras have zero weight
  // (m=2 band shifted to kb 10..19 so all LDS reads stay in range).
  for (int t = wave; t < 15; t += 8) {       // 3 M-tiles x 5 N-tiles
    const int m   = t / 5;
    const int nt  = t - m * 5;
    const int kb0 = (m == 1) ? 7 : ((m == 2) ? 10 : 0);
    const int o   = 16 * m + lrow;
    const int nc  = nt * 16 + lrow;
    v8f acc = {};
#pragma unroll
    for (int kk = 0; kk < 10; ++kk) {
      const int kb = kb0 + kk;
      v2f a, b;
#pragma unroll
      for (int j = 0; j < 2; ++j) {
        const int k = kb * 4 + lhalf * 2 + j;
        a[j] = whval(o, k);
        b[j] = sIn[k][nc];
      }
      acc = __builtin_amdgcn_wmma_f32_16x16x4_f32(false, a, false, b,
                                                  (short)0, acc, false, false);
    }
#pragma unroll
    for (int r = 0; r < 8; ++r)
      sMid[16 * m + lhalf * 8 + r][nc] = acc[r];
  }
  __syncthreads();

  // ---- stage 2: out[48][32] = sMid[48][80] @ WwT[80][32], banded K ----
  // Output local col n' taps LDS mid cols 2n'+7 .. 2n'+10 -> 10 K-steps.
  for (int t = wave; t < 6; t += 8) {        // 3 M-tiles x 2 N-tiles
    const int m  = t >> 1;
    const int nt = t & 1;
    const int M  = 16 * m + lrow;            // A: M = lane&15 within tile
    const int np = 16 * nt + lrow;           // B: N = lane&15 within tile
    v8f acc = {};
#pragma unroll
    for (int kk = 0; kk < 10; ++kk) {
      const int kb = 8 * nt + 1 + kk;
      v2f a, b;
#pragma unroll
      for (int j = 0; j < 2; ++j) {
        const int k = kb * 4 + lhalf * 2 + j;
        a[j] = sMid[M][k];
        b[j] = wsel(k - (2 * np + 7));
      }
      acc = __builtin_amdgcn_wmma_f32_16x16x4_f32(false, a, false, b,
                                                  (short)0, acc, false, false);
    }
    // D layout: row = r + 8*(lane>>4), col = lane&15. Scale, clamp, store.
#pragma unroll
    for (int r = 0; r < 8; ++r) {
      const int row = 16 * m + lhalf * 8 + r;
      if (row < OUT_H) {
        const int col = strip * 32 + 16 * nt + lrow;
        float v = acc[r] * (1.0f / 255.0f);
        v = fminf(fmaxf(v, 0.0f), 1.0f);
        out[((size_t)(box * 3 + ch) * OUT_H + row) * OUT_W + col] = v;
      }
    }
  }
}

extern "C" void kernel_launch(void* const* d_in, const int* in_sizes, int n_in,
                              void* d_out, int out_size, void* d_ws, size_t ws_size,
                              hipStream_t stream) {
  const float* img   = (const float*)d_in[0];   // [8,1080,1920,3] fp32 NHWC
  const int*   boxes = (const int*)d_in[1];     // [256,5] int32
  float*       out   = (float*)d_out;           // [256,3,40,320] fp32

  dim3 grid(30, 256);   // (strip 0..9 x channel 0..2, box)
  dim3 block(256);      // 8 wave32s
  hipLaunchKernelGGL(crop_resize_bicubic_wmma, grid, block, 0, stream,
                     img, boxes, out);
}